// GlobalAttentionModule_7438883356930
// MI455X (gfx1250) — compile-verified
//
#include <hip/hip_runtime.h>

// out = relu(GroupNorm32(Wv @ feat + bv))   -- exact algebraic reduction of the
// reference: softmax over the last axis sums to 1, and v is broadcast over that
// axis, so the whole pairwise/attention pipeline is an identity on v.

#define C_DIM 128
#define N_DIM 512
#define B_DIM 2
#define EPS_F 1e-5f

typedef __attribute__((ext_vector_type(2))) float v2f;
typedef __attribute__((ext_vector_type(8))) float v8f;

// -------- Kernel 1: vpre[b,o,n] = sum_c Wv[o,c]*feat[b,c,n] + bv[o] ---------
// One wave32 computes a 16(M) x 16(N) f32 tile with 32 chained
// V_WMMA_F32_16X16X4_F32 steps (K = 128). Full f32 precision.
//
// ISA layouts (CDNA5 ISA):
//   A (16x4 MxK, 2 VGPRs): lane l, vgpr r -> A[m = l&15][k = 2*(l>>4) + r]
//   B (4x16 KxN, 2 VGPRs): lane l, vgpr r -> B[k = 2*(l>>4) + r][n = l&15]
//   C/D (16x16, 8 VGPRs) : lane l, vgpr r -> D[m = r + 8*(l>>4)][n = l&15]
__global__ __launch_bounds__(128) void vproj_wmma_f32(
    const float* __restrict__ feat, const float* __restrict__ Wv,
    const float* __restrict__ bv, float* __restrict__ vpre)
{
  const int wave = blockIdx.x * (blockDim.x >> 5) + (threadIdx.x >> 5);
  const int lane = threadIdx.x & 31;
  // 2 batches * 8 M-tiles * 32 N-tiles = 512 waves total
  const int b    = wave >> 8;
  const int mt   = (wave >> 5) & 7;
  const int nt   = wave & 31;
  const int m0   = mt << 4;
  const int n0   = nt << 4;

  const int half = lane >> 4;   // lane half selects K sub-pair
  const int l15  = lane & 15;
  const int kb   = half << 1;

  const float* __restrict__ featb = feat + (size_t)b * C_DIM * N_DIM;
  const float* __restrict__ arow  = Wv + (size_t)(m0 + l15) * C_DIM + kb;
  const float* __restrict__ bcol  = featb + n0 + l15;

  v8f acc = {};
  #pragma unroll 4
  for (int kk = 0; kk < C_DIM; kk += 4) {
    // A operand: contiguous float2 from the Wv row
    v2f a = *(const v2f*)(arow + kk);
    // B operand: two strided elements of the feat tile column
    v2f bm;
    bm.x = bcol[(size_t)(kk + kb    ) * N_DIM];
    bm.y = bcol[(size_t)(kk + kb + 1) * N_DIM];
    // 8 args: (neg_a, A, neg_b, B, c_mod, C, reuse_a, reuse_b)
    acc = __builtin_amdgcn_wmma_f32_16x16x4_f32(false, a, false, bm,
                                                (short)0, acc, false, false);
  }

  #pragma unroll
  for (int r = 0; r < 8; ++r) {
    const int mo = m0 + r + (half << 3);
    vpre[((size_t)b * C_DIM + mo) * N_DIM + n0 + l15] = acc[r] + bv[mo];
  }
}

// -------- Kernel 2: GroupNorm(32 groups over C=128) + affine + ReLU ---------
// One block per (batch, group): 4 channels x 512 = 2048 elements.
__global__ __launch_bounds__(256) void gn_relu(
    const float* __restrict__ vpre, const float* __restrict__ gamma,
    const float* __restrict__ beta, float* __restrict__ out)
{
  const int b = blockIdx.x >> 5;   // 64 blocks = 2 batches * 32 groups
  const int g = blockIdx.x & 31;
  const float* __restrict__ src = vpre + ((size_t)b * C_DIM + g * 4) * N_DIM;
  float* __restrict__ dst       = out  + ((size_t)b * C_DIM + g * 4) * N_DIM;

  float x[8];
  float s = 0.f, sq = 0.f;
  #pragma unroll
  for (int i = 0; i < 8; ++i) {
    x[i] = src[threadIdx.x + i * 256];
    s  += x[i];
    sq += x[i] * x[i];
  }
  // intra-wave reduction (wave32)
  #pragma unroll
  for (int off = 16; off > 0; off >>= 1) {
    s  += __shfl_xor(s, off, 32);
    sq += __shfl_xor(sq, off, 32);
  }
  __shared__ float red_s[8];
  __shared__ float red_q[8];
  const int w = threadIdx.x >> 5;
  if ((threadIdx.x & 31) == 0) { red_s[w] = s; red_q[w] = sq; }
  __syncthreads();
  float ts = 0.f, tq = 0.f;
  #pragma unroll
  for (int i = 0; i < 8; ++i) { ts += red_s[i]; tq += red_q[i]; }

  const float mean = ts * (1.f / 2048.f);
  const float var  = tq * (1.f / 2048.f) - mean * mean;  // biased, as jnp.var
  const float inv  = rsqrtf(var + EPS_F);

  #pragma unroll
  for (int i = 0; i < 8; ++i) {
    const int idx = threadIdx.x + i * 256;
    const int ch  = g * 4 + (idx >> 9);                   // idx / 512
    const float y = (x[i] - mean) * inv * gamma[ch] + beta[ch];
    dst[idx] = fmaxf(y, 0.f);
  }
}

extern "C" void kernel_launch(void* const* d_in, const int* in_sizes, int n_in,
                              void* d_out, int out_size, void* d_ws, size_t ws_size,
                              hipStream_t stream) {
  // setup_inputs order:
  // 0 feat, 1 Wk, 2 bk, 3 Wq, 4 bq, 5 Wv, 6 bv, 7 gn_v_g, 8 gn_v_b,
  // 9 gn1_g, 10 gn1_b, 11 W1, 12 b1, 13 gn2_g, 14 gn2_b, 15 W2, 16 b2
  const float* feat = (const float*)d_in[0];
  const float* Wv   = (const float*)d_in[5];
  const float* bv   = (const float*)d_in[6];
  const float* gg   = (const float*)d_in[7];
  const float* gb   = (const float*)d_in[8];

  float* vpre = (float*)d_ws;            // B*C*N f32 = 512 KB scratch
  float* outp = (float*)d_out;           // B*C*N f32

  vproj_wmma_f32<<<128, 128, 0, stream>>>(feat, Wv, bv, vpre);
  gn_relu<<<64, 256, 0, stream>>>(vpre, gg, gb, outp);
}